// ToRGB_3917010174310
// MI455X (gfx1250) — compile-verified
//
#include <hip/hip_runtime.h>

typedef float v2f __attribute__((ext_vector_type(2)));
typedef float v8f __attribute__((ext_vector_type(8)));

#define C_CH 512
#define SD 512
#define BATCH 16
#define H_OUT 128
#define W_OUT 128
#define HW (H_OUT * W_OUT)
#define H_SK 64
#define W_SK 64

// ---------------------------------------------------------------------------
// Kernel 1: modulation GEMM via WMMA f32 16x16x4.
//   s[b][ch] = sum_k style[b][k] * mod_weight[ch][k] * (1/sqrt(SD)) + mod_bias[ch]
//   wmod[b][o][ch] = conv_weight[o][ch] * (1/sqrt(C)) * s[b][ch]
// One wave (32 threads) per 16-channel tile; M=16 == batch dimension.
// ---------------------------------------------------------------------------
__global__ __launch_bounds__(32) void modulate_wmma_kernel(
    const float* __restrict__ style,       // [16,512]
    const float* __restrict__ mod_weight,  // [512,512]
    const float* __restrict__ mod_bias,    // [512]
    const float* __restrict__ conv_weight, // [3,512]
    float* __restrict__ wmod)              // [16,3,512]
{
    const float inv_sqrt_sd = 0.04419417382415922f; // 1/sqrt(512)
    const float inv_sqrt_c  = 0.04419417382415922f; // 1/sqrt(512)

    const int lane  = threadIdx.x;        // 0..31, wave32
    const int row   = lane & 15;          // M (lanes 0-15) / N row index
    const int khalf = lane >> 4;          // 0: K=0,1   1: K=2,3
    const int n0    = blockIdx.x * 16;    // channel tile base

    v8f acc = {};                         // 16x16 f32 accumulator (8 VGPRs)

    for (int k0 = 0; k0 < SD; k0 += 4) {
        const int ka = k0 + khalf * 2;
        // A (16x4, M=batch): lane row holds style[row][ka], style[row][ka+1]
        v2f a;
        a.x = style[row * SD + ka];
        a.y = style[row * SD + ka + 1];
        // B (4x16, N=channel): lane row holds mod_weight[n0+row][ka..ka+1],
        // with the 1/sqrt(SD) scale folded in.
        v2f b;
        b.x = mod_weight[(n0 + row) * SD + ka] * inv_sqrt_sd;
        b.y = mod_weight[(n0 + row) * SD + ka + 1] * inv_sqrt_sd;
        acc = __builtin_amdgcn_wmma_f32_16x16x4_f32(
            /*neg_a=*/false, a, /*neg_b=*/false, b,
            /*c_mod=*/(short)0, acc, /*reuse_a=*/false, /*reuse_b=*/false);
    }

    // D layout: VGPR r -> (M = r + 8*khalf, N = lane&15)
    const int ch  = n0 + row;
    const float mb  = mod_bias[ch];
    const float cw0 = conv_weight[0 * C_CH + ch] * inv_sqrt_c;
    const float cw1 = conv_weight[1 * C_CH + ch] * inv_sqrt_c;
    const float cw2 = conv_weight[2 * C_CH + ch] * inv_sqrt_c;

#pragma unroll
    for (int r = 0; r < 8; ++r) {
        const int bb = r + khalf * 8;     // batch index
        const float s = acc[r] + mb;
        wmod[(bb * 3 + 0) * C_CH + ch] = cw0 * s;
        wmod[(bb * 3 + 1) * C_CH + ch] = cw1 * s;
        wmod[(bb * 3 + 2) * C_CH + ch] = cw2 * s;
    }
}

// ---------------------------------------------------------------------------
// Kernel 2: bandwidth-bound streaming contraction over 512 channels + fused
// bias add + FIR 2x upsample of skip. Each thread owns 4 consecutive pixels
// (one float4 per channel iteration -> 512B/wave fully coalesced loads).
// ---------------------------------------------------------------------------
__device__ __forceinline__ float skip_at(const float* __restrict__ sk,
                                         int o, int iy, int ix) {
    if ((unsigned)iy < (unsigned)H_SK && (unsigned)ix < (unsigned)W_SK)
        return sk[(o * H_SK + iy) * W_SK + ix];
    return 0.0f;
}

__global__ __launch_bounds__(256) void torgb_stream_kernel(
    const float* __restrict__ x,     // [16,512,128,128]
    const float* __restrict__ wmod,  // [16,3,512]
    const float* __restrict__ bias,  // [3]
    const float* __restrict__ skip,  // [16,3,64,64]
    float* __restrict__ out)         // [16,3,128,128]
{
    const int b    = blockIdx.y;
    const int pix0 = blockIdx.x * 1024 + threadIdx.x * 4;

    const float4* __restrict__ xp =
        reinterpret_cast<const float4*>(x + (size_t)b * C_CH * HW + pix0);
    const float* __restrict__ wb = wmod + b * 3 * C_CH;

    float a0x = 0.f, a0y = 0.f, a0z = 0.f, a0w = 0.f;
    float a1x = 0.f, a1y = 0.f, a1z = 0.f, a1w = 0.f;
    float a2x = 0.f, a2y = 0.f, a2z = 0.f, a2w = 0.f;

#pragma unroll 4
    for (int i = 0; i < C_CH; ++i) {
        const float4 xv = xp[(size_t)i * (HW / 4)];
        const float w0 = wb[i];              // block-uniform -> scalar loads
        const float w1 = wb[C_CH + i];
        const float w2 = wb[2 * C_CH + i];
        a0x = fmaf(w0, xv.x, a0x); a0y = fmaf(w0, xv.y, a0y);
        a0z = fmaf(w0, xv.z, a0z); a0w = fmaf(w0, xv.w, a0w);
        a1x = fmaf(w1, xv.x, a1x); a1y = fmaf(w1, xv.y, a1y);
        a1z = fmaf(w1, xv.z, a1z); a1w = fmaf(w1, xv.w, a1w);
        a2x = fmaf(w2, xv.x, a2x); a2y = fmaf(w2, xv.y, a2y);
        a2z = fmaf(w2, xv.z, a2z); a2w = fmaf(w2, xv.w, a2w);
    }

    float acc[3][4] = {{a0x, a0y, a0z, a0w},
                       {a1x, a1y, a1z, a1w},
                       {a2x, a2y, a2z, a2w}};

    // ---- fused epilogue: bias + separable FIR 2x upsample of skip ----
    const float* __restrict__ skb = skip + (size_t)b * 3 * H_SK * W_SK;
    const int y  = pix0 / W_OUT;             // all 4 pixels share one row
    const int x0 = pix0 % W_OUT;

    int iy_a; float wy_a, wy_b;
    if (y & 1) { iy_a = y >> 1;       wy_a = 0.75f; wy_b = 0.25f; }
    else       { iy_a = (y >> 1) - 1; wy_a = 0.25f; wy_b = 0.75f; }
    const int iy_b = iy_a + 1;

    const float bz[3] = {bias[0], bias[1], bias[2]};

#pragma unroll
    for (int j = 0; j < 4; ++j) {
        const int xx = x0 + j;
        int ix_a; float wx_a, wx_b;
        if (xx & 1) { ix_a = xx >> 1;       wx_a = 0.75f; wx_b = 0.25f; }
        else        { ix_a = (xx >> 1) - 1; wx_a = 0.25f; wx_b = 0.75f; }
        const int ix_b = ix_a + 1;
#pragma unroll
        for (int o = 0; o < 3; ++o) {
            const float fir =
                wy_a * (wx_a * skip_at(skb, o, iy_a, ix_a) +
                        wx_b * skip_at(skb, o, iy_a, ix_b)) +
                wy_b * (wx_a * skip_at(skb, o, iy_b, ix_a) +
                        wx_b * skip_at(skb, o, iy_b, ix_b));
            acc[o][j] += bz[o] + fir;
        }
    }

#pragma unroll
    for (int o = 0; o < 3; ++o) {
        float4 v = make_float4(acc[o][0], acc[o][1], acc[o][2], acc[o][3]);
        *reinterpret_cast<float4*>(out + ((size_t)(b * 3 + o)) * HW + pix0) = v;
    }
}

extern "C" void kernel_launch(void* const* d_in, const int* in_sizes, int n_in,
                              void* d_out, int out_size, void* d_ws, size_t ws_size,
                              hipStream_t stream) {
    const float* x           = (const float*)d_in[0]; // [16,512,128,128]
    const float* style       = (const float*)d_in[1]; // [16,512]
    const float* skip        = (const float*)d_in[2]; // [16,3,64,64]
    const float* conv_weight = (const float*)d_in[3]; // [3,512]
    const float* mod_weight  = (const float*)d_in[4]; // [512,512]
    const float* mod_bias    = (const float*)d_in[5]; // [512]
    const float* bias        = (const float*)d_in[6]; // [3]
    float* out  = (float*)d_out;
    float* wmod = (float*)d_ws;                       // 16*3*512 floats = 96 KiB

    // Kernel 1: WMMA modulation GEMM -> per-batch modulated 3x512 weights.
    modulate_wmma_kernel<<<dim3(SD / 16), dim3(32), 0, stream>>>(
        style, mod_weight, mod_bias, conv_weight, wmod);

    // Kernel 2: bandwidth-bound contraction + fused bias/FIR upsample.
    dim3 grid(HW / 1024, BATCH);
    torgb_stream_kernel<<<grid, dim3(256), 0, stream>>>(
        x, wmod, bias, skip, out);
}